// SinkhornIterationLayer_16913581211845
// MI455X (gfx1250) — compile-verified
//
#include <hip/hip_runtime.h>
#include <stdint.h>

// Sinkhorn iteration for B=128 matrices of 256x256 fp32.
// Strategy (MI455X / gfx1250):
//   - One workgroup (1024 threads = 32 wave32) per matrix.
//   - Whole matrix (256 KB) lives in LDS (WGP has 320 KB) for all <=15 iterations.
//   - HBM traffic = read 32MB + write 32MB total (roofline ~2.7us @ 23.3 TB/s),
//     instead of ~1.9 GB for a streaming multi-kernel formulation.
//   - Global<->LDS copies use CDNA5 async-to/from-LDS (ASYNCcnt) path: no VGPR
//     round trip for the bulk data. Copy loops are deliberately NOT unrolled so
//     only one address triple is live at a time (avoids VGPR spills seen when
//     all 16 async ops had their addresses hoisted).
//   - y = x * rcpU[col] * rcpV[row]; reductions are LDS/bank-conflict-free,
//     cross-lane via wave32 __shfl_xor.

#define MAT_N       256
#define MAT_ELEMS   (MAT_N * MAT_N)      // 65536
#define NTHREADS    1024
#define NWAVES      (NTHREADS / 32)      // 32
#define SK_MAX_ITER 15
#define SK_EPS2     1e-12f               // (1e-6)^2, squared Frobenius threshold

// LDS layout (floats):
//   mat  [65536]  matrix, updated in place
//   rcpU [256]    1/colsum   (0 where colsum==0, matching divide_no_nan)
//   rcpV [256]    1/rowsum(h)
//   colp [4096]   16 row-phases x 256 columns of partial column sums
//   red  [33]     per-wave diff partials + final total
#define SMEM_FLOATS (MAT_ELEMS + MAT_N + MAT_N + 16 * MAT_N + NWAVES + 1)

__global__ __launch_bounds__(NTHREADS)
void sinkhorn_iter_kernel(const float* __restrict__ xin, float* __restrict__ yout)
{
    extern __shared__ float smem[];
    float* mat  = smem;
    float* rcpU = smem + MAT_ELEMS;
    float* rcpV = rcpU + MAT_N;
    float* colp = rcpV + MAT_N;
    float* red  = colp + 16 * MAT_N;

    const int tid  = threadIdx.x;
    const int lane = tid & 31;
    const int wave = tid >> 5;
    const int ph   = tid >> 6;    // row phase 0..15 (rows ph, 16+ph, ..., 240+ph)
    const int cg   = tid & 63;    // column group: columns 4*cg .. 4*cg+3

    const float* gx = xin  + (size_t)blockIdx.x * MAT_ELEMS;
    float*       gy = yout + (size_t)blockIdx.x * MAT_ELEMS;

    float4*       mat4  = (float4*)mat;
    const float4* rcpU4 = (const float4*)rcpU;

    // ---------------- load matrix into LDS (async global -> LDS) ----------------
#if defined(__gfx1250__)
    {
        // Low 32 bits of a generic pointer into LDS are the LDS byte address
        // (LDS aperture: addr[31:0] is the wave-relative LDS offset).
        uint32_t ldsAddr = (uint32_t)(uintptr_t)mat + (uint32_t)tid * 16u;
        const char* gaddr = (const char*)gx + (uint32_t)tid * 16u;
        #pragma unroll 1
        for (int k = 0; k < 16; ++k) {
            asm volatile("global_load_async_to_lds_b128 %0, %1, off"
                         :: "v"(ldsAddr), "v"(gaddr)
                         : "memory");
            ldsAddr += NTHREADS * 16u;
            gaddr   += NTHREADS * 16u;
        }
#if __has_builtin(__builtin_amdgcn_s_wait_asynccnt)
        __builtin_amdgcn_s_wait_asynccnt(0);
#else
        asm volatile("s_wait_asynccnt 0" ::: "memory");
#endif
    }
#else
    #pragma unroll
    for (int k = 0; k < 16; ++k) {
        int i4 = k * NTHREADS + tid;
        mat4[i4] = ((const float4*)gx)[i4];
    }
#endif
    __syncthreads();

    for (int iter = 0; iter < SK_MAX_ITER; ++iter) {
        // ---- pass 1: column sums u[c] ----
        // Thread owns columns 4*cg..4*cg+3 for its 16 rows {16k + ph}.
        {
            float4 acc = make_float4(0.f, 0.f, 0.f, 0.f);
            #pragma unroll
            for (int k = 0; k < 16; ++k) {
                float4 v = mat4[k * NTHREADS + tid];
                acc.x += v.x; acc.y += v.y; acc.z += v.z; acc.w += v.w;
            }
            ((float4*)colp)[ph * 64 + cg] = acc;   // colp[ph][4*cg..+3]
        }
        __syncthreads();
        if (tid < MAT_N) {
            float u = 0.0f;
            #pragma unroll
            for (int p = 0; p < 16; ++p) u += colp[p * MAT_N + tid];
            rcpU[tid] = (u == 0.0f) ? 0.0f : 1.0f / u;   // divide_no_nan
        }
        __syncthreads();

        // ---- pass 2: row sums of h = x * rcpU; one wave per row ----
        {
            float4 u0 = rcpU4[lane];
            float4 u1 = rcpU4[32 + lane];
            #pragma unroll
            for (int j = 0; j < 8; ++j) {
                int r = wave * 8 + j;
                float4 a0 = mat4[r * 64 + lane];
                float4 a1 = mat4[r * 64 + 32 + lane];
                float acc = a0.x * u0.x + a0.y * u0.y + a0.z * u0.z + a0.w * u0.w
                          + a1.x * u1.x + a1.y * u1.y + a1.z * u1.z + a1.w * u1.w;
                #pragma unroll
                for (int off = 16; off > 0; off >>= 1)
                    acc += __shfl_xor(acc, off, 32);
                if (lane == 0)
                    rcpV[r] = (acc == 0.0f) ? 0.0f : 1.0f / acc;   // divide_no_nan
            }
        }
        __syncthreads();

        // ---- pass 3: y = x * rcpU[c] * rcpV[r], in place; accumulate ||y-x||^2 ----
        float dsum = 0.0f;
        {
            float4 ru = rcpU4[cg];              // thread's column group is k-invariant
            #pragma unroll
            for (int k = 0; k < 16; ++k) {
                int   i4 = k * NTHREADS + tid;
                float rv = rcpV[k * 16 + ph];   // wave-uniform -> LDS broadcast
                float4 xv = mat4[i4];
                float4 yv;
                yv.x = (xv.x * ru.x) * rv;
                yv.y = (xv.y * ru.y) * rv;
                yv.z = (xv.z * ru.z) * rv;
                yv.w = (xv.w * ru.w) * rv;
                float dx = yv.x - xv.x;
                float dy = yv.y - xv.y;
                float dz = yv.z - xv.z;
                float dw = yv.w - xv.w;
                dsum += dx * dx + dy * dy + dz * dz + dw * dw;
                mat4[i4] = yv;
            }
        }

        // ---- block-wide convergence reduction ----
        #pragma unroll
        for (int off = 16; off > 0; off >>= 1)
            dsum += __shfl_xor(dsum, off, 32);
        if (lane == 0) red[wave] = dsum;
        __syncthreads();
        if (wave == 0) {
            float t = red[lane];                // NWAVES == 32
            #pragma unroll
            for (int off = 16; off > 0; off >>= 1)
                t += __shfl_xor(t, off, 32);
            if (lane == 0) red[NWAVES] = t;
        }
        __syncthreads();
        if (red[NWAVES] < SK_EPS2) break;       // uniform across the block
    }

    // ---------------- write result back (async LDS -> global) ----------------
#if defined(__gfx1250__)
    {
        // All waves' matrix updates are LDS-visible (last __syncthreads above).
        // Drain this wave's own pending DS stores before the async engine reads LDS.
#if __has_builtin(__builtin_amdgcn_s_wait_dscnt)
        __builtin_amdgcn_s_wait_dscnt(0);
#else
        asm volatile("s_wait_dscnt 0" ::: "memory");
#endif
        uint32_t ldsAddr = (uint32_t)(uintptr_t)mat + (uint32_t)tid * 16u;
        char* gaddr = (char*)gy + (uint32_t)tid * 16u;
        #pragma unroll 1
        for (int k = 0; k < 16; ++k) {
            asm volatile("global_store_async_from_lds_b128 %0, %1, off"
                         :: "v"(gaddr), "v"(ldsAddr)
                         : "memory");
            ldsAddr += NTHREADS * 16u;
            gaddr   += NTHREADS * 16u;
        }
#if __has_builtin(__builtin_amdgcn_s_wait_asynccnt)
        __builtin_amdgcn_s_wait_asynccnt(0);
#else
        asm volatile("s_wait_asynccnt 0" ::: "memory");
#endif
    }
#else
    #pragma unroll
    for (int k = 0; k < 16; ++k) {
        int i4 = k * NTHREADS + tid;
        ((float4*)gy)[i4] = mat4[i4];
    }
#endif
}

extern "C" void kernel_launch(void* const* d_in, const int* in_sizes, int n_in,
                              void* d_out, int out_size, void* d_ws, size_t ws_size,
                              hipStream_t stream)
{
    (void)n_in; (void)out_size; (void)d_ws; (void)ws_size;
    const float* x = (const float*)d_in[0];
    float*       y = (float*)d_out;
    const int batch = in_sizes[0] / MAT_ELEMS;          // 128 matrices
    const size_t smemBytes = (size_t)SMEM_FLOATS * sizeof(float);  // ~274 KB < 320 KB/WGP

    // Allow >64KB dynamic LDS (CDNA5 WGP has 320 KB). Deterministic, capture-safe.
    hipFuncSetAttribute((const void*)sinkhorn_iter_kernel,
                        hipFuncAttributeMaxDynamicSharedMemorySize, (int)smemBytes);

    sinkhorn_iter_kernel<<<batch, NTHREADS, smemBytes, stream>>>(x, y);
}